// EvalModel_19859928776820
// MI455X (gfx1250) — compile-verified
//
#include <hip/hip_runtime.h>
#include <hip/hip_bf16.h>

typedef __attribute__((ext_vector_type(2))) float v2f;
typedef __attribute__((ext_vector_type(8))) float v8f;

#define HID 64
#define G4  256   // 4*HID

// ---------------------------------------------------------------------------
// Phase A: xW1 = x @ W1 + b1   (x: [T,2], W1: [2,256]) -- trivially parallel
// ---------------------------------------------------------------------------
__global__ __launch_bounds__(256) void xw1_kernel(const float* __restrict__ x,
                                                  const float* __restrict__ W1,
                                                  const float* __restrict__ b1,
                                                  float* __restrict__ xw, int T) {
    int idx = blockIdx.x * 256 + threadIdx.x;      // T*256 total
    if (idx >= T * G4) return;
    int t = idx >> 8, j = idx & 255;
    float v = fmaf(x[2 * t], W1[j], b1[j]);
    v = fmaf(x[2 * t + 1], W1[G4 + j], v);
    xw[idx] = v;
}

// ---------------------------------------------------------------------------
// Phase B: xW = A @ W + b  where A = previous layer's h sequence [T,64],
// W: [64,256].  f32 WMMA (V_WMMA_F32_16X16X4_F32), exact f32 semantics.
// One 16x16 output tile per wave; 16 k-steps of K=4 cover K=64.
// Grid: (T/16, 2), block 256 (8 waves -> 8 column tiles; 2 y-blocks -> 16).
// ---------------------------------------------------------------------------
__global__ __launch_bounds__(256) void xw_gemm_wmma(const float* __restrict__ A,
                                                    const float* __restrict__ W,
                                                    const float* __restrict__ bias,
                                                    float* __restrict__ out, int T) {
    const int wave = threadIdx.x >> 5;                 // 0..7
    const int lane = threadIdx.x & 31;
    const int t0   = blockIdx.x * 16;                  // row tile
    const int n0   = ((blockIdx.y << 3) + wave) * 16;  // col tile
    if (t0 >= T) return;

    const int r  = lane & 15;    // row index (A/D) and column index (B/D)
    const int hi = lane >> 4;    // half-wave selector

    v8f acc = {};
    const float* Arow = A + (size_t)(t0 + r) * HID;    // this lane's A row

    #pragma unroll
    for (int kk = 0; kk < 16; ++kk) {
        const int kb = kk * 4 + hi * 2;                // K pair this half-wave owns
        // A-matrix 16x4 f32 frag: lane r holds A[r][kb], A[r][kb+1] (contiguous)
        v2f a;
        a.x = Arow[kb];
        a.y = Arow[kb + 1];
        // B-matrix 4x16 f32 frag: VGPR v -> K=kb+v, N = n0 + r
        v2f b;
        b.x = W[(size_t)kb * G4 + n0 + r];
        b.y = W[(size_t)(kb + 1) * G4 + n0 + r];
        acc = __builtin_amdgcn_wmma_f32_16x16x4_f32(
            /*neg_a=*/false, a, /*neg_b=*/false, b,
            /*c_mod=*/(short)0, acc, /*reuse_a=*/false, /*reuse_b=*/false);
    }

    const float bn = bias[n0 + r];
    #pragma unroll
    for (int v = 0; v < 8; ++v) {
        const int m = v + hi * 8;                      // C/D layout: M = v + 8*hi
        out[(size_t)(t0 + m) * G4 + n0 + r] = acc[v] + bn;
    }
}

// ---------------------------------------------------------------------------
// Phase C: sequential LSTM scan.  ONE persistent 256-thread block (8 wave32).
// Thread j owns gate column j; its U column (64 floats) lives in VGPRs.
// h broadcast via LDS (ds_load_b128), c state in registers of threads 0..63.
// 64 FMAs/lane/step with 4 independent accumulators + 2 barriers per step.
// ---------------------------------------------------------------------------
__global__ __launch_bounds__(256) void lstm_scan(const float* __restrict__ xW,
                                                 const float* __restrict__ U,
                                                 float* __restrict__ h_seq,
                                                 float* __restrict__ h_final,
                                                 int T, int store_seq) {
    __shared__ __align__(16) float hs[HID];
    __shared__ float act[G4];
    const int j = threadIdx.x;

    float u[HID];
    #pragma unroll
    for (int k = 0; k < HID; ++k) u[k] = U[(size_t)k * G4 + j];

    if (j < HID) hs[j] = 0.0f;
    float c = 0.0f;
    __syncthreads();

    for (int t = 0; t < T; ++t) {
        float z = xW[(size_t)t * G4 + j];
        if (t + 4 < T) __builtin_prefetch(&xW[(size_t)(t + 4) * G4 + j], 0, 0);

        float z0 = 0.f, z1 = 0.f, z2 = 0.f, z3 = 0.f;
        #pragma unroll
        for (int k = 0; k < HID / 4; ++k) {
            float4 hv = ((const float4*)hs)[k];        // ds_load_b128 broadcast
            z0 = fmaf(hv.x, u[4 * k + 0], z0);
            z1 = fmaf(hv.y, u[4 * k + 1], z1);
            z2 = fmaf(hv.z, u[4 * k + 2], z2);
            z3 = fmaf(hv.w, u[4 * k + 3], z3);
        }
        z += (z0 + z1) + (z2 + z3);

        float a;
        if ((j >> 6) == 2) a = tanhf(z);               // g gate (cols 128..191)
        else               a = 1.0f / (1.0f + __expf(-z)); // i, f, o gates
        act[j] = a;
        __syncthreads();

        if (j < HID) {
            float ig = act[j];
            float fg = act[HID + j];
            float gg = act[2 * HID + j];
            float og = act[3 * HID + j];
            c = fmaf(fg, c, ig * gg);
            float h = og * tanhf(c);
            hs[j] = h;
            if (store_seq)            h_seq[(size_t)t * HID + j] = h;
            else if (t == T - 1)      h_final[j] = h;
        }
        __syncthreads();
    }
}

// ---------------------------------------------------------------------------
// Phase D: MLP head on the final hidden state (tiny; one block).
// ---------------------------------------------------------------------------
__global__ __launch_bounds__(64) void head_kernel(const float* __restrict__ hfin,
                                                  const float* __restrict__ Wd1,
                                                  const float* __restrict__ bd1,
                                                  const float* __restrict__ Wd2,
                                                  const float* __restrict__ bd2,
                                                  const float* __restrict__ Wl,
                                                  const float* __restrict__ bl,
                                                  float* __restrict__ out) {
    __shared__ float h4[20], h6[20];
    const int j = threadIdx.x;
    if (j < 20) {
        float s = bd1[j];
        #pragma unroll
        for (int k = 0; k < HID; ++k) s = fmaf(hfin[k], Wd1[k * 20 + j], s);
        h4[j] = fmaxf(s, 0.0f);
    }
    __syncthreads();
    if (j < 20) {
        float s = bd2[j];
        #pragma unroll
        for (int k = 0; k < 20; ++k) s = fmaf(h4[k], Wd2[k * 20 + j], s);
        h6[j] = fmaxf(s, 0.0f);
    }
    __syncthreads();
    if (j < 10) {
        float s = bl[j];
        #pragma unroll
        for (int k = 0; k < 20; ++k) s = fmaf(h6[k], Wl[k * 10 + j], s);
        out[j] = s;
    }
}

// ---------------------------------------------------------------------------
extern "C" void kernel_launch(void* const* d_in, const int* in_sizes, int n_in,
                              void* d_out, int out_size, void* d_ws, size_t ws_size,
                              hipStream_t stream) {
    const float* x   = (const float*)d_in[0];
    const float* W1  = (const float*)d_in[1];
    const float* U1  = (const float*)d_in[2];
    const float* b1  = (const float*)d_in[3];
    const float* W2  = (const float*)d_in[4];
    const float* U2  = (const float*)d_in[5];
    const float* b2  = (const float*)d_in[6];
    const float* W3  = (const float*)d_in[7];
    const float* U3  = (const float*)d_in[8];
    const float* b3  = (const float*)d_in[9];
    const float* Wd1 = (const float*)d_in[10];
    const float* bd1 = (const float*)d_in[11];
    const float* Wd2 = (const float*)d_in[12];
    const float* bd2 = (const float*)d_in[13];
    const float* Wl  = (const float*)d_in[14];
    const float* bl  = (const float*)d_in[15];
    float* out = (float*)d_out;

    const int T = in_sizes[0] / 2;                     // x is [T,2]

    // Workspace layout (reused across layers; all resident in 192MB L2):
    float* xw   = (float*)d_ws;                        // [T,256]
    float* h    = xw + (size_t)T * G4;                 // [T,64] (h1 then h2)
    float* hfin = h + (size_t)T * HID;                 // [64]

    // Layer 1
    xw1_kernel<<<T, 256, 0, stream>>>(x, W1, b1, xw, T);
    lstm_scan<<<1, 256, 0, stream>>>(xw, U1, h, hfin, T, 1);
    // Layer 2: input projection as a real GEMM over the whole sequence (WMMA)
    xw_gemm_wmma<<<dim3(T / 16, 2), 256, 0, stream>>>(h, W2, b2, xw, T);
    lstm_scan<<<1, 256, 0, stream>>>(xw, U2, h, hfin, T, 1);
    // Layer 3
    xw_gemm_wmma<<<dim3(T / 16, 2), 256, 0, stream>>>(h, W3, b3, xw, T);
    lstm_scan<<<1, 256, 0, stream>>>(xw, U3, h, hfin, T, 0);
    // Head
    head_kernel<<<1, 64, 0, stream>>>(hfin, Wd1, bd1, Wd2, bd2, Wl, bl, out);
}